// AttnBlock_34213709480336
// MI455X (gfx1250) — compile-verified
//
#include <hip/hip_runtime.h>
#include <math.h>

// ---------------------------------------------------------------------------
// HypergraphConv attention forward for MI455X (gfx1250, wave32, WMMA).
//   N=50000 nodes, M=10000 edges, E=500000 incidences, IN_CH=256, H=8, F=32.
// Dense GEMMs -> fp32 WMMA (V_WMMA_F32_16X16X4_F32, exact fp32 math),
//   A strip staged in LDS (ds_load_b64 fragments), full-tile fast-path store.
// Attention logits decomposed: alpha = leaky(s_n[node] + s_e[edge]).
// Segment softmax via int-trick float atomicMax + atomicAdd (L2 atomics).
// Message passes: vectorized gathers (L2-resident) + global_atomic_add_f32.
// ---------------------------------------------------------------------------

#define HEADS 8
#define OUTC 32
#define CH 256                  // IN_CH == HEADS*OUTC
#define NEG_SLOPE 0.2f
#define EPSI 1e-16f

typedef __attribute__((ext_vector_type(2))) float v2f;
typedef __attribute__((ext_vector_type(8))) float v8f;

__global__ void fill_kernel(float* __restrict__ p, float v, long n) {
    long i = (long)blockIdx.x * blockDim.x + threadIdx.x;
    if (i < n) p[i] = v;
}

// Y[rows,256] = X[rows,256] @ W[256,256] with V_WMMA_F32_16X16X4_F32.
// 16 waves/block; wave w owns the 16x16 C tile at columns [16w, 16w+16).
// The 16x256 A strip is staged once in LDS and shared by all 16 waves.
// fp32 A 16x4 fragment layout (ISA 7.12.2): lanes 0-15 hold K=0(v0),K=1(v1);
// lanes 16-31 hold K=2(v0),K=3(v1). B 4x16 mirrors it. C/D: VGPR r ->
// row r (lanes 0-15) / row r+8 (lanes 16-31).
__global__ __launch_bounds__(512)
void gemm_wmma_f32(const float* __restrict__ X, const float* __restrict__ W,
                   float* __restrict__ Y, int rows) {
    __shared__ float As[16 * CH];                // 16 KB strip

    const int tid  = threadIdx.x;
    const int lane = tid & 31;
    const int wave = tid >> 5;                   // 0..15 -> column tile
    const int row0 = blockIdx.x * 16;

    // Cooperative stage of A strip: 4096 floats = 1024 float4, 512 threads.
    #pragma unroll
    for (int v = 0; v < 2; ++v) {
        int f4  = v * 512 + tid;                 // float4 index 0..1023
        int idx = f4 << 2;                       // float index
        int r   = idx >> 8;                      // strip row 0..15
        int cc  = idx & 255;                     // channel
        int gr  = row0 + r;
        if (gr >= rows) gr = rows - 1;           // clamp reads (remainder tile)
        *(float4*)(As + idx) = *(const float4*)(X + (long)gr * CH + cc);
    }
    __syncthreads();

    const int col0  = wave * 16;
    const int l15   = lane & 15;
    const int khalf = (lane >> 4) << 1;          // 0 or 2
    const float* __restrict__ Ar = As + l15 * CH;      // LDS row for this lane
    const float* __restrict__ Wp = W + col0 + l15;     // column for this lane

    v8f c = {};
    #pragma unroll 4
    for (int k0 = 0; k0 < CH; k0 += 4) {
        v2f a, b;
        a.x = Ar[k0 + khalf + 0];                // ds_load_b64 pair
        a.y = Ar[k0 + khalf + 1];
        b.x = Wp[(long)(k0 + khalf + 0) * CH];
        b.y = Wp[(long)(k0 + khalf + 1) * CH];
        c = __builtin_amdgcn_wmma_f32_16x16x4_f32(false, a, false, b,
                                                  (short)0, c, false, false);
    }

    const int rbase = row0 + ((lane >> 4) << 3);
    float* __restrict__ Yp = Y + (long)rbase * CH + col0 + l15;
    if (row0 + 16 <= rows) {                     // full tile: branch-free stores
        #pragma unroll
        for (int r = 0; r < 8; ++r)
            Yp[(long)r * CH] = c[r];
    } else {                                     // remainder tile: guarded
        #pragma unroll
        for (int r = 0; r < 8; ++r)
            if (rbase + r < rows) Yp[(long)r * CH] = c[r];
    }
}

// S[r,h] = sum_f F[r, h*32+f] * att[h, att_off+f]
__global__ void scores_kernel(const float* __restrict__ Fm,
                              const float* __restrict__ att, int att_off,
                              float* __restrict__ S, int rows) {
    int i = blockIdx.x * blockDim.x + threadIdx.x;
    if (i >= rows * HEADS) return;
    int r = i >> 3, h = i & 7;
    const float* f = Fm + (long)r * CH + h * OUTC;
    const float* a = att + h * (2 * OUTC) + att_off;
    float s = 0.f;
    #pragma unroll
    for (int j = 0; j < OUTC; ++j) s += f[j] * a[j];
    S[i] = s;
}

// Float atomic-max via signed/unsigned int ordering trick (native L2 atomics).
__device__ __forceinline__ void atomicMaxF(float* addr, float v) {
    if (v >= 0.f) atomicMax((int*)addr, __float_as_int(v));
    else atomicMin((unsigned int*)addr, (unsigned int)__float_as_int(v));
}

// Per incidence: logits + leaky relu + segment max + degree counts.
__global__ void logits_kernel(const int* __restrict__ node, const int* __restrict__ edge,
                              const float* __restrict__ sn, const float* __restrict__ se,
                              float* __restrict__ alpha, float* __restrict__ amax,
                              float* __restrict__ Dc, float* __restrict__ Bc, int E) {
    int e = blockIdx.x * blockDim.x + threadIdx.x;
    if (e >= E) return;
    int n = node[e], m = edge[e];
    atomicAdd(&Dc[n], 1.f);
    atomicAdd(&Bc[m], 1.f);
    #pragma unroll
    for (int h = 0; h < HEADS; ++h) {
        float l = sn[n * HEADS + h] + se[m * HEADS + h];
        l = (l > 0.f) ? l : NEG_SLOPE * l;
        alpha[(long)e * HEADS + h] = l;
        atomicMaxF(&amax[n * HEADS + h], l);
    }
}

__global__ void expsum_kernel(const int* __restrict__ node, float* __restrict__ alpha,
                              const float* __restrict__ amax, float* __restrict__ asum,
                              long EH) {
    long i = (long)blockIdx.x * blockDim.x + threadIdx.x;
    if (i >= EH) return;
    long e = i >> 3; int h = (int)(i & 7);
    int n = node[e];
    float a = expf(alpha[i] - amax[n * HEADS + h]);
    alpha[i] = a;
    atomicAdd(&asum[n * HEADS + h], a);
}

__global__ void norm_kernel(const int* __restrict__ node, float* __restrict__ alpha,
                            const float* __restrict__ asum, long EH) {
    long i = (long)blockIdx.x * blockDim.x + threadIdx.x;
    if (i >= EH) return;
    long e = i >> 3; int h = (int)(i & 7);
    alpha[i] = alpha[i] / (asum[node[e] * HEADS + h] + EPSI);
}

__global__ void invert_kernel(float* __restrict__ p, int n) {
    int i = blockIdx.x * blockDim.x + threadIdx.x;
    if (i >= n) return;
    float v = p[i];
    p[i] = (v > 0.f) ? 1.f / v : 0.f;
}

// Pass 1: nodes -> hyperedges. One thread per (incidence, 4 channels).
__global__ void pass1_kernel(const int* __restrict__ node, const int* __restrict__ edge,
                             const float* __restrict__ alpha, const float* __restrict__ Binv,
                             const float* __restrict__ xh, float* __restrict__ out_e,
                             long total) {
    long i = (long)blockIdx.x * blockDim.x + threadIdx.x;
    if (i >= total) return;                 // total = E*64
    long e = i >> 6;
    int c4 = (int)(i & 63) << 2;            // channel 0..252 step 4
    int h = c4 >> 5;
    int n = node[e], m = edge[e];
    float coef = Binv[m] * alpha[e * HEADS + h];
    const float4 xv = *(const float4*)(xh + (long)n * CH + c4);
    float* dst = out_e + (long)m * CH + c4;
    atomicAdd(dst + 0, coef * xv.x);
    atomicAdd(dst + 1, coef * xv.y);
    atomicAdd(dst + 2, coef * xv.z);
    atomicAdd(dst + 3, coef * xv.w);
}

// Pass 2: hyperedges -> nodes.
__global__ void pass2_kernel(const int* __restrict__ node, const int* __restrict__ edge,
                             const float* __restrict__ alpha, const float* __restrict__ Dinv,
                             const float* __restrict__ out_e, float* __restrict__ out_n,
                             long total) {
    long i = (long)blockIdx.x * blockDim.x + threadIdx.x;
    if (i >= total) return;                 // total = E*64
    long e = i >> 6;
    int c4 = (int)(i & 63) << 2;
    int h = c4 >> 5;
    int n = node[e], m = edge[e];
    float coef = Dinv[n] * alpha[e * HEADS + h];
    const float4 ev = *(const float4*)(out_e + (long)m * CH + c4);
    float* dst = out_n + (long)n * CH + c4;
    atomicAdd(dst + 0, coef * ev.x);
    atomicAdd(dst + 1, coef * ev.y);
    atomicAdd(dst + 2, coef * ev.z);
    atomicAdd(dst + 3, coef * ev.w);
}

extern "C" void kernel_launch(void* const* d_in, const int* in_sizes, int n_in,
                              void* d_out, int out_size, void* d_ws, size_t ws_size,
                              hipStream_t stream) {
    const float* x    = (const float*)d_in[0];
    const int*   hidx = (const int*)  d_in[1];
    const float* hea  = (const float*)d_in[2];
    const float* W    = (const float*)d_in[3];
    const float* att  = (const float*)d_in[4];

    const int N = in_sizes[0] / CH;
    const int E = in_sizes[1] / 2;
    const int M = in_sizes[2] / CH;
    const int* node = hidx;          // hyperedge_index[0]
    const int* edge = hidx + E;      // hyperedge_index[1]

    // Workspace carve-out (floats). ~93 MB total for the given sizes.
    float* ws   = (float*)d_ws;
    float* xh   = ws;                      size_t off = (size_t)N * CH;
    float* eh   = ws + off;                off += (size_t)M * CH;
    float* sn   = ws + off;                off += (size_t)N * HEADS;
    float* se   = ws + off;                off += (size_t)M * HEADS;
    float* amax = ws + off;                off += (size_t)N * HEADS;
    float* asum = ws + off;                off += (size_t)N * HEADS;
    float* Dc   = ws + off;                off += (size_t)N;
    float* Bc   = ws + off;                off += (size_t)M;
    float* alpha= ws + off;                off += (size_t)E * HEADS;
    float* oute = ws + off;                off += (size_t)M * CH;
    (void)ws_size; (void)n_in;

    float* out  = (float*)d_out;
    const int T = 256;
    auto blk = [](long n, int t) { return (unsigned)((n + t - 1) / t); };

    // Init accumulators (harness poisons buffers; must be deterministic).
    fill_kernel<<<blk((long)N * CH, T), T, 0, stream>>>(out,  0.f, (long)N * CH);
    fill_kernel<<<blk((long)M * CH, T), T, 0, stream>>>(oute, 0.f, (long)M * CH);
    fill_kernel<<<blk((long)N * HEADS, T), T, 0, stream>>>(amax, -3.402823466e38f, (long)N * HEADS);
    fill_kernel<<<blk((long)N * HEADS, T), T, 0, stream>>>(asum, 0.f, (long)N * HEADS);
    fill_kernel<<<blk(N, T), T, 0, stream>>>(Dc, 0.f, N);
    fill_kernel<<<blk(M, T), T, 0, stream>>>(Bc, 0.f, M);

    // Dense projections on the matrix pipe (fp32 WMMA, LDS-staged A strip).
    gemm_wmma_f32<<<(N + 15) / 16, 512, 0, stream>>>(x,   W, xh, N);
    gemm_wmma_f32<<<(M + 15) / 16, 512, 0, stream>>>(hea, W, eh, M);

    // Decomposed attention scores.
    scores_kernel<<<blk((long)N * HEADS, T), T, 0, stream>>>(xh, att, 0,    sn, N);
    scores_kernel<<<blk((long)M * HEADS, T), T, 0, stream>>>(eh, att, OUTC, se, M);

    // Logits + leaky relu + segment max + degrees.
    logits_kernel<<<blk(E, T), T, 0, stream>>>(node, edge, sn, se, alpha, amax, Dc, Bc, E);

    // Segment softmax.
    const long EH = (long)E * HEADS;
    expsum_kernel<<<blk(EH, T), T, 0, stream>>>(node, alpha, amax, asum, EH);
    norm_kernel  <<<blk(EH, T), T, 0, stream>>>(node, alpha, asum, EH);

    // Degree inverses in place.
    invert_kernel<<<blk(N, T), T, 0, stream>>>(Dc, N);
    invert_kernel<<<blk(M, T), T, 0, stream>>>(Bc, M);

    // Two scatter passes (L2-resident gathers + f32 atomics).
    const long tot = (long)E * 64;
    pass1_kernel<<<blk(tot, T), T, 0, stream>>>(node, edge, alpha, Bc, xh, oute, tot);
    pass2_kernel<<<blk(tot, T), T, 0, stream>>>(node, edge, alpha, Dc, oute, out, tot);
}